// NodeEdgeBlock_5463198401131
// MI455X (gfx1250) — compile-verified
//
#include <hip/hip_runtime.h>
#include <hip/hip_bf16.h>
#include <math.h>
#include <stdint.h>

// ---------------------------------------------------------------------------
// NodeEdgeBlock for MI455X (gfx1250, wave32, WMMA).
// Strategy: one workgroup per (b, i) holds the full 256x256 f32 Y slice in
// LDS (256KB + padding, within the 320KB/WGP budget), so the 268MB Y tensor
// never touches HBM. All GEMMs use V_WMMA_F32_16X16X4_F32 (fp32-exact).
// E-chunks are streamed into LDS with the gfx1250 async global->LDS path
// (GLOBAL_LOAD_ASYNC_TO_LDS_B128 + S_WAIT_ASYNCCNT), double-buffered so the
// HBM latency of the only streaming input overlaps the WMMA work.
// ---------------------------------------------------------------------------

#define BS 4
#define NN 256
#define DX 256
#define DE 64
#define DY 64
#define NH 8
#define DF 32

typedef float v2f __attribute__((ext_vector_type(2)));
typedef float v8f __attribute__((ext_vector_type(8)));

__device__ __forceinline__ v8f wmma4(v2f a, v2f b, v8f c) {
  // D = A(16x4,f32) * B(4x16,f32) + C(16x16,f32)
  return __builtin_amdgcn_wmma_f32_16x16x4_f32(
      /*neg_a=*/false, a, /*neg_b=*/false, b,
      /*c_mod=*/(short)0, c, /*reuse_a=*/false, /*reuse_b=*/false);
}

// LDS aperture base is 2^32-aligned (ISA 10.2: isLDS checks ADDR[63:32] only),
// so the low 32 bits of a flat shared pointer are the LDS-relative address.
__device__ __forceinline__ unsigned lds_off(const void* p) {
  return (unsigned)(uintptr_t)p;
}

// Async global->LDS 16-byte copy (per-lane), tracked by ASYNCcnt.
__device__ __forceinline__ void async_ld16(unsigned lds_byte_off, const float* g) {
  asm volatile("global_load_async_to_lds_b128 %0, %1, off"
               :: "v"(lds_byte_off), "v"(g)
               : "memory");
}
__device__ __forceinline__ void wait_async0() {
  asm volatile("s_wait_asynccnt 0x0" ::: "memory");
}

// ---------------------------------------------------------------------------
// Kernel 1: per-batch gating vectors from y:  ye_add, ye_mul, yx_add, yx_mul
// ---------------------------------------------------------------------------
__global__ void gates_kernel(const float* __restrict__ y,
                             const float* Wyea, const float* byea,
                             const float* Wyem, const float* byem,
                             const float* Wyxa, const float* byxa,
                             const float* Wyxm, const float* byxm,
                             float* ye_add, float* ye_mul,
                             float* yx_add, float* yx_mul) {
  int b = blockIdx.x, c = threadIdx.x;
  const float* yb = y + b * DY;
  float s0 = byea[c], s1 = byem[c], s2 = byxa[c], s3 = byxm[c];
  for (int k = 0; k < DY; ++k) {
    float yk = yb[k];
    s0 += yk * Wyea[k * DX + c];
    s1 += yk * Wyem[k * DX + c];
    s2 += yk * Wyxa[k * DX + c];
    s3 += yk * Wyxm[k * DX + c];
  }
  ye_add[b * DX + c] = s0;
  ye_mul[b * DX + c] = s1;
  yx_add[b * DX + c] = s2;
  yx_mul[b * DX + c] = s3;
}

// ---------------------------------------------------------------------------
// Kernel 2: Q/K/V = (X @ W + b) * mask   (WMMA f32 16x16x4)
// grid = (N/64, 3, BS), block = 256 (8 waves). Each wave owns 8 output tiles.
// ---------------------------------------------------------------------------
__global__ void qkv_kernel(const float* __restrict__ X,
                           const float* Wq, const float* bq,
                           const float* Wk, const float* bk,
                           const float* Wv, const float* bv,
                           const float* __restrict__ mask,
                           float* Q, float* K, float* V) {
  int b = blockIdx.z;
  int which = blockIdx.y;
  int rowbase = blockIdx.x * 64;
  const float* W    = (which == 0) ? Wq : (which == 1) ? Wk : Wv;
  const float* bias = (which == 0) ? bq : (which == 1) ? bk : bv;
  float* Out        = (which == 0) ? Q  : (which == 1) ? K  : V;
  const float* Xb = X + (size_t)b * NN * DX;

  int wave = threadIdx.x >> 5;
  int lane = threadIdx.x & 31;
  int lr = lane & 15, kh = lane >> 4;

  for (int t8 = wave; t8 < 64; t8 += 8) {
    int mt = t8 & 3, nt = t8 >> 2;
    int row0 = rowbase + mt * 16;
    v8f acc = {0.f, 0.f, 0.f, 0.f, 0.f, 0.f, 0.f, 0.f};
    for (int k = 0; k < DX; k += 4) {
      v2f a, bf;
      const float* ap = Xb + (size_t)(row0 + lr) * DX + k + 2 * kh;
      a.x = ap[0]; a.y = ap[1];
      const float* bp = W + (size_t)(k + 2 * kh) * DX + nt * 16 + lr;
      bf.x = bp[0]; bf.y = bp[DX];
      acc = wmma4(a, bf, acc);
    }
    int col = nt * 16 + lr;
    float bc = bias[col];
    for (int v = 0; v < 8; ++v) {
      int r = row0 + v + 8 * kh;
      Out[((size_t)b * NN + r) * DX + col] = (acc[v] + bc) * mask[b * NN + r];
    }
  }
}

// ---------------------------------------------------------------------------
// Kernel 3: X statistics -> zx (mean,min,max,std[ddof=1]) per (b,c)
// ---------------------------------------------------------------------------
__global__ void xstats_kernel(const float* __restrict__ X, float* zx) {
  int b = blockIdx.x, c = threadIdx.x;
  float s = 0.f, sq = 0.f, mn = INFINITY, mx = -INFINITY;
  for (int i = 0; i < NN; ++i) {
    float v = X[((size_t)b * NN + i) * DX + c];
    s += v; sq += v * v; mn = fminf(mn, v); mx = fmaxf(mx, v);
  }
  float mean = s / NN;
  float var = (sq - s * s / NN) / (NN - 1);
  float* z = zx + (size_t)b * 4 * DX;
  z[c] = mean; z[DX + c] = mn; z[2 * DX + c] = mx;
  z[3 * DX + c] = sqrtf(fmaxf(var, 0.f));
}

// ---------------------------------------------------------------------------
// Kernel 4: E statistics, stage 1 (partial sums per (b, part, c))
// grid = (16, BS), block 256 (4 row-groups x 64 channels) -> coalesced reads.
// ---------------------------------------------------------------------------
__global__ void estats_kernel(const float* __restrict__ E, float* part) {
  int b = blockIdx.y, p = blockIdx.x;
  int t = threadIdx.x, rr = t >> 6, c = t & 63;
  float s = 0.f, sq = 0.f, mn = INFINITY, mx = -INFINITY;
  const float* Eb = E + (size_t)b * NN * NN * DE;
  for (int row = p * 4096 + rr; row < (p + 1) * 4096; row += 4) {
    float v = Eb[(size_t)row * DE + c];
    s += v; sq += v * v; mn = fminf(mn, v); mx = fmaxf(mx, v);
  }
  __shared__ float red[4][256];
  red[0][t] = s; red[1][t] = sq; red[2][t] = mn; red[3][t] = mx;
  __syncthreads();
  if (rr == 0) {
    float S  = red[0][c] + red[0][64 + c] + red[0][128 + c] + red[0][192 + c];
    float SQ = red[1][c] + red[1][64 + c] + red[1][128 + c] + red[1][192 + c];
    float MN = fminf(fminf(red[2][c], red[2][64 + c]),
                     fminf(red[2][128 + c], red[2][192 + c]));
    float MX = fmaxf(fmaxf(red[3][c], red[3][64 + c]),
                     fmaxf(red[3][128 + c], red[3][192 + c]));
    float* o = part + ((size_t)(b * 16 + p) * DE + c) * 4;
    o[0] = S; o[1] = SQ; o[2] = MN; o[3] = MX;
  }
}

// ---------------------------------------------------------------------------
// Kernel 5: finalize ze, then the small y MLP tail -> new_y
// 1 block, 256 threads: thread t = (b, c) with b=t/64, c=t%64.
// ---------------------------------------------------------------------------
__global__ void yfinal_kernel(const float* __restrict__ y,
                              const float* __restrict__ zx,
                              const float* __restrict__ part,
                              const float* Wyy, const float* byy,
                              const float* Wxy, const float* bxy,
                              const float* Wey, const float* bey,
                              const float* W1, const float* b1,
                              const float* W2, const float* b2,
                              float* out_y) {
  __shared__ float zeL[BS][4 * DE];
  __shared__ float yn[BS][DY];
  __shared__ float h[BS][DY];
  int t = threadIdx.x, b = t >> 6, c = t & 63;

  float S = 0.f, SQ = 0.f, MN = INFINITY, MX = -INFINITY;
  for (int p = 0; p < 16; ++p) {
    const float* o = part + ((size_t)(b * 16 + p) * DE + c) * 4;
    S += o[0]; SQ += o[1]; MN = fminf(MN, o[2]); MX = fmaxf(MX, o[3]);
  }
  const float cnt = 65536.f;
  zeL[b][c]          = S / cnt;
  zeL[b][DE + c]     = MN;
  zeL[b][2 * DE + c] = MX;
  zeL[b][3 * DE + c] = sqrtf(fmaxf((SQ - S * S / cnt) / (cnt - 1.f), 0.f));
  __syncthreads();

  float acc = byy[c] + bxy[c] + bey[c];
  for (int k = 0; k < DY; ++k)     acc += y[b * DY + k] * Wyy[k * DY + c];
  for (int k = 0; k < 4 * DX; ++k) acc += zx[(size_t)b * 4 * DX + k] * Wxy[k * DY + c];
  for (int k = 0; k < 4 * DE; ++k) acc += zeL[b][k] * Wey[k * DY + c];
  yn[b][c] = acc;
  __syncthreads();

  float a2 = b1[c];
  for (int k = 0; k < DY; ++k) a2 += yn[b][k] * W1[k * DY + c];
  h[b][c] = fmaxf(a2, 0.f);
  __syncthreads();

  float a3 = b2[c];
  for (int k = 0; k < DY; ++k) a3 += h[b][k] * W2[k * DY + c];
  out_y[b * DY + c] = a3;
}

// ---------------------------------------------------------------------------
// Kernel 6: fused edge pipeline + channelwise attention, one block per (b,i).
// LDS holds the whole Y[b,i,:,:] slice (stride 257 -> bank-conflict-free
// WMMA A-fragment reads). E chunks double-buffered via async global->LDS.
// ---------------------------------------------------------------------------
__global__ __launch_bounds__(256) void edge_kernel(
    const float* __restrict__ E,
    const float* __restrict__ Qw, const float* __restrict__ Kw,
    const float* __restrict__ Vw, const float* __restrict__ mask,
    const float* __restrict__ Wem, const float* __restrict__ bem,
    const float* __restrict__ Wea, const float* __restrict__ bea,
    const float* __restrict__ ye_add, const float* __restrict__ ye_mul,
    const float* __restrict__ yx_add, const float* __restrict__ yx_mul,
    const float* __restrict__ Weo, const float* __restrict__ beo,
    const float* __restrict__ Wxo, const float* __restrict__ bxo,
    float* __restrict__ newX, float* __restrict__ newE) {
  extern __shared__ float lds[];
  const int YS = 257;                 // padded row stride for Y
  float* Y    = lds;                  // 256 * 257 = 65792
  float* EaS  = lds + 65792;          // 32 * 256  =  8192
  float* Ech0 = EaS + 8192;           // 32 * 65   =  2080 (padded)
  float* Ech1 = Ech0 + 2080;          // 32 * 65   =  2080 (double buffer)
  float* qv   = Ech1 + 2080;          // 256
  float* mj   = qv + 256;             // 256
  float* wv   = mj + 256;             // 256  (total 78912 floats = 308.3 KB)

  int b = blockIdx.y, i = blockIdx.x;
  int t = threadIdx.x;
  int wave = t >> 5, lane = t & 31, lr = lane & 15, kh = lane >> 4;
  float mi = mask[b * NN + i];

  qv[t] = Qw[((size_t)b * NN + i) * DX + t];
  mj[t] = mask[b * NN + t];

  const unsigned ech_off[2] = {lds_off(Ech0), lds_off(Ech1)};
  const float* Erow = E + ((size_t)(b * NN + i)) * NN * DE;

  // Prefetch chunk 0 (512 x b128; this thread issues idx = t, t+256).
  // Chunk rows land at padded stride 65 so WMMA A-reads are conflict-free.
  for (int idx = t; idx < 512; idx += 256) {
    int row = idx >> 4, c4 = (idx & 15) << 2;
    async_ld16(ech_off[0] + (unsigned)((row * 65 + c4) * 4),
               Erow + row * 64 + c4);
  }
  __syncthreads();  // covers qv/mj staging

  const float rsdf = 0.17677669529663687f;  // 1/sqrt(DF=32)

  for (int jc = 0; jc < 8; ++jc) {
    wait_async0();      // this thread's async copies for chunk jc are in LDS
    __syncthreads();    // -> everyone's copies are in LDS & visible

    // Prefetch chunk jc+1 into the other buffer (readers of that buffer
    // finished before the previous iteration's closing barrier).
    if (jc < 7) {
      const float* Enext = Erow + (size_t)(jc + 1) * 32 * DE;
      for (int idx = t; idx < 512; idx += 256) {
        int row = idx >> 4, c4 = (idx & 15) << 2;
        async_ld16(ech_off[(jc + 1) & 1] + (unsigned)((row * 65 + c4) * 4),
                   Enext + row * 64 + c4);
      }
    }
    const float* Ech = (jc & 1) ? Ech1 : Ech0;

    // --- WMMA: Em chunk -> Y rows (raw), Ea chunk -> EaS (raw) ---
    for (int t8 = wave; t8 < 32; t8 += 8) {
      int mt = t8 & 1, nt = t8 >> 1;
      v8f am = {0.f, 0.f, 0.f, 0.f, 0.f, 0.f, 0.f, 0.f};
      v8f aa = {0.f, 0.f, 0.f, 0.f, 0.f, 0.f, 0.f, 0.f};
      for (int k = 0; k < DE; k += 4) {
        v2f af;
        const float* ap = Ech + (mt * 16 + lr) * 65 + k + 2 * kh;
        af.x = ap[0]; af.y = ap[1];
        v2f bm, ba;
        const float* bp1 = Wem + (size_t)(k + 2 * kh) * DX + nt * 16 + lr;
        bm.x = bp1[0]; bm.y = bp1[DX];
        const float* bp2 = Wea + (size_t)(k + 2 * kh) * DX + nt * 16 + lr;
        ba.x = bp2[0]; ba.y = bp2[DX];
        am = wmma4(af, bm, am);
        aa = wmma4(af, ba, aa);
      }
      int col = nt * 16 + lr;
      for (int v = 0; v < 8; ++v) {
        int rloc = mt * 16 + v + 8 * kh;  // 0..31 within chunk
        Y[(jc * 32 + rloc) * YS + col] = am[v];
        EaS[rloc * 256 + col] = aa[v];
      }
    }
    __syncthreads();

    // --- combine: Y = (Q.K/sqrt(df)) * (E1+1) + E2, with masks/biases ---
    {
      int c = t;
      float bemc = bem[c], beac = bea[c], q = qv[c];
      for (int jj = 0; jj < 32; ++jj) {
        int j = jc * 32 + jj;
        float mm = mi * mj[j];
        float em = (Y[j * YS + c] + bemc) * mm;
        float ea = (EaS[jj * 256 + c] + beac) * mm;
        float kv = Kw[((size_t)b * NN + j) * DX + c];
        Y[j * YS + c] = q * kv * rsdf * (em + 1.f) + ea;
      }
    }
    __syncthreads();
  }

  // --- channelwise masked softmax over keys j, weighted V, yx gating ---
  {
    int c = t;
    float m = -INFINITY;
    for (int j = 0; j < NN; ++j)
      if (mj[j] > 0.f) m = fmaxf(m, Y[j * YS + c]);
    float s = 0.f, wsum = 0.f;
    for (int j = 0; j < NN; ++j)
      if (mj[j] > 0.f) {
        float e = __expf(Y[j * YS + c] - m);
        s += e;
        wsum += e * Vw[((size_t)b * NN + j) * DX + c];
      }
    float weighted = wsum / s;
    wv[c] = yx_add[b * DX + c] + (yx_mul[b * DX + c] + 1.f) * weighted;
  }
  __syncthreads();

  // --- newX = (wv @ W_xout + b) * mask_i ---
  {
    int c = t;
    float acc = bxo[c];
    for (int k = 0; k < DX; ++k) acc += wv[k] * Wxo[(size_t)k * DX + c];
    newX[((size_t)b * NN + i) * DX + c] = acc * mi;
  }

  // --- ye gating in place (column c touched only by thread c: no barrier) ---
  {
    int c = t;
    float g1 = ye_add[b * DX + c], g2 = ye_mul[b * DX + c] + 1.f;
    for (int j = 0; j < NN; ++j) Y[j * YS + c] = g1 + g2 * Y[j * YS + c];
  }
  __syncthreads();

  // --- WMMA: newE = (gatedY @ W_eout + b) * mask_i * mask_j ---
  for (int t8 = wave; t8 < 64; t8 += 8) {
    int mt = t8 & 15, nt = t8 >> 4;
    v8f acc = {0.f, 0.f, 0.f, 0.f, 0.f, 0.f, 0.f, 0.f};
    for (int k = 0; k < DX; k += 4) {
      v2f af;
      const float* ap = Y + (mt * 16 + lr) * YS + k + 2 * kh;
      af.x = ap[0]; af.y = ap[1];
      v2f bf;
      const float* bp = Weo + (size_t)(k + 2 * kh) * DE + nt * 16 + lr;
      bf.x = bp[0]; bf.y = bp[DE];
      acc = wmma4(af, bf, acc);
    }
    int cd = nt * 16 + lr;
    float bc = beo[cd];
    for (int v = 0; v < 8; ++v) {
      int j = mt * 16 + v + 8 * kh;
      newE[(((size_t)(b * NN + i)) * NN + j) * DE + cd] =
          (acc[v] + bc) * mi * mj[j];
    }
  }
}

// ---------------------------------------------------------------------------
// Host launch
// ---------------------------------------------------------------------------
extern "C" void kernel_launch(void* const* d_in, const int* in_sizes, int n_in,
                              void* d_out, int out_size, void* d_ws,
                              size_t ws_size, hipStream_t stream) {
  const float* X    = (const float*)d_in[0];
  const float* E    = (const float*)d_in[1];
  const float* y    = (const float*)d_in[2];
  const float* mask = (const float*)d_in[3];
  // params flattened in _LAYERS order, W then b per layer:
  const float* Wq   = (const float*)d_in[4];   const float* bq   = (const float*)d_in[5];
  const float* Wk   = (const float*)d_in[6];   const float* bk   = (const float*)d_in[7];
  const float* Wv   = (const float*)d_in[8];   const float* bv   = (const float*)d_in[9];
  const float* Wea  = (const float*)d_in[10];  const float* bea  = (const float*)d_in[11];  // e_add
  const float* Wem  = (const float*)d_in[12];  const float* bem  = (const float*)d_in[13];  // e_mul
  const float* Wyem = (const float*)d_in[14];  const float* byem = (const float*)d_in[15];  // y_e_mul
  const float* Wyea = (const float*)d_in[16];  const float* byea = (const float*)d_in[17];  // y_e_add
  const float* Wyxm = (const float*)d_in[18];  const float* byxm = (const float*)d_in[19];  // y_x_mul
  const float* Wyxa = (const float*)d_in[20];  const float* byxa = (const float*)d_in[21];  // y_x_add
  const float* Wyy  = (const float*)d_in[22];  const float* byy  = (const float*)d_in[23];
  const float* Wxy  = (const float*)d_in[24];  const float* bxy  = (const float*)d_in[25];
  const float* Wey  = (const float*)d_in[26];  const float* bey  = (const float*)d_in[27];
  const float* Wxo  = (const float*)d_in[28];  const float* bxo  = (const float*)d_in[29];  // x_out
  const float* Weo  = (const float*)d_in[30];  const float* beo  = (const float*)d_in[31];  // e_out
  const float* W1   = (const float*)d_in[32];  const float* b1   = (const float*)d_in[33];  // y_out1
  const float* W2   = (const float*)d_in[34];  const float* b2   = (const float*)d_in[35];  // y_out2

  float* out = (float*)d_out;
  float* newX  = out;                        // 4*256*256          = 262144
  float* newE  = out + 262144;               // 4*256*256*256      = 67108864
  float* new_y = out + 262144 + 67108864;    // 4*64

  float* ws = (float*)d_ws;
  float* Qb     = ws;             // 262144
  float* Kb     = ws + 262144;    // 262144
  float* Vb     = ws + 524288;    // 262144
  float* ye_add = ws + 786432;    // 1024
  float* ye_mul = ws + 787456;    // 1024
  float* yx_add = ws + 788480;    // 1024
  float* yx_mul = ws + 789504;    // 1024
  float* zx     = ws + 790528;    // 4096
  float* epart  = ws + 794624;    // 16384
  (void)ws_size; (void)n_in; (void)in_sizes; (void)out_size;

  gates_kernel<<<BS, 256, 0, stream>>>(y, Wyea, byea, Wyem, byem,
                                       Wyxa, byxa, Wyxm, byxm,
                                       ye_add, ye_mul, yx_add, yx_mul);

  qkv_kernel<<<dim3(NN / 64, 3, BS), 256, 0, stream>>>(
      X, Wq, bq, Wk, bk, Wv, bv, mask, Qb, Kb, Vb);

  xstats_kernel<<<BS, 256, 0, stream>>>(X, zx);
  estats_kernel<<<dim3(16, BS), 256, 0, stream>>>(E, epart);
  yfinal_kernel<<<1, 256, 0, stream>>>(y, zx, epart, Wyy, byy, Wxy, bxy,
                                       Wey, bey, W1, b1, W2, b2, new_y);

  const size_t edge_lds = (size_t)78912 * sizeof(float);  // ~308 KB of 320 KB
  edge_kernel<<<dim3(NN, BS), 256, edge_lds, stream>>>(
      E, Qb, Kb, Vb, mask, Wem, bem, Wea, bea,
      ye_add, ye_mul, yx_add, yx_mul, Weo, beo, Wxo, bxo, newX, newE);
}